// QuantumFilterLayer_87428354277458
// MI455X (gfx1250) — compile-verified
//
#include <hip/hip_runtime.h>

// cos(x + w) over [B=4194304, NQ=24] f32. Touch-once streaming elementwise op:
// 805 MB total traffic -> ~35us floor at 23.3 TB/s HBM. Native v_cos_f32
// (v_mul by 1/2pi + v_cos per element) keeps compute far below the memory
// roof; |x+w| < ~12 rad ~= 2 revolutions, inside v_cos_f32's valid range.
//
// Layout trick: one thread processes one full 24-float row (6 x b128 chunks)
// per grid-stride iteration. Then the weight pattern is simply w[0..23] for
// every thread: no modulo, no LDS, no barrier. The 24 weight loads are
// uniform (constant indices, const __restrict__, never written) so they
// scalarize to s_load -> weights sit in SGPRs and feed v_add_f32 directly.
// Steady-state loop: clause of 6 global_load_b128 (NT) -> one wait ->
// 24x(v_add, v_mul, v_cos) -> clause of 6 global_store_b128 (NT).

typedef float v4f __attribute__((ext_vector_type(4), aligned(16)));

#define NQ 24
#define CPR 6        // chunks per row: 24 floats / 4
#define THREADS 256  // 8 wave32s

__global__ void __launch_bounds__(THREADS)
qfl_cos_kernel(const float* __restrict__ x,
               const float* __restrict__ w,
               float* __restrict__ out,
               int nRows, int tailStart, int nTotal)
{
    // Uniform weights -> SGPRs (s_load), used as scalar VALU operands.
    float wr[NQ];
    #pragma unroll
    for (int i = 0; i < NQ; ++i) wr[i] = w[i];

    const v4f* __restrict__ xv = (const v4f*)x;
    v4f*       __restrict__ ov = (v4f*)out;

    int idx    = blockIdx.x * THREADS + threadIdx.x;
    int stride = gridDim.x * THREADS;

    for (int r = idx; r < nRows; r += stride) {
        int base = r * CPR;

        v4f v[CPR];
        #pragma unroll
        for (int j = 0; j < CPR; ++j)            // clause: 6x global_load_b128 (NT)
            v[j] = __builtin_nontemporal_load(xv + base + j);

        #pragma unroll
        for (int j = 0; j < CPR; ++j) {
            v4f rr;
            rr.x = __cosf(v[j].x + wr[4*j + 0]); // v_add, v_mul(1/2pi), v_cos_f32
            rr.y = __cosf(v[j].y + wr[4*j + 1]);
            rr.z = __cosf(v[j].z + wr[4*j + 2]);
            rr.w = __cosf(v[j].w + wr[4*j + 3]);
            __builtin_nontemporal_store(rr, ov + base + j); // global_store_b128 (NT)
        }
    }

    // Tail guard for flat lengths not divisible by 24 (unused for this shape:
    // at most 23 scalar elements).
    if (idx == 0) {
        for (int i = tailStart; i < nTotal; ++i)
            out[i] = __cosf(x[i] + wr[i % NQ]);
    }
}

extern "C" void kernel_launch(void* const* d_in, const int* in_sizes, int n_in,
                              void* d_out, int out_size, void* d_ws, size_t ws_size,
                              hipStream_t stream)
{
    const float* x = (const float*)d_in[0];   // [B, 24] f32
    const float* w = (const float*)d_in[1];   // [24]    f32
    float* out     = (float*)d_out;           // [B, 24] f32

    int n         = in_sizes[0];
    int nRows     = n / (NQ);                 // full 24-float rows
    int tailStart = nRows * NQ;

    int blocks = (nRows + THREADS - 1) / THREADS;
    // 8192 blocks * 256 threads = 2.1M threads -> ~2 rows (384 B) per thread:
    // enough in-flight b128 requests to saturate HBM, minimal setup overhead.
    if (blocks > 8192) blocks = 8192;
    if (blocks < 1) blocks = 1;

    qfl_cos_kernel<<<blocks, THREADS, 0, stream>>>(x, w, out, nRows, tailStart, n);
}